// Randomized_Rotation_Quasi_SWD_987842478822
// MI455X (gfx1250) — compile-verified
//
#include <hip/hip_runtime.h>
#include <math.h>

#define BATCH 32
#define NPTS  4096
#define NPROJ 256

typedef __attribute__((ext_vector_type(2))) float v2f;
typedef __attribute__((ext_vector_type(8))) float v8f;

// ---------------------------------------------------------------------------
// Kernel 1: proj[b][p][e] = sum_d theta[p][d] * rotations[b][d][e]
// ---------------------------------------------------------------------------
__global__ void proj_kernel(const float* __restrict__ theta,
                            const float* __restrict__ rot,
                            float* __restrict__ proj) {
    const int b = blockIdx.x;
    const int p = threadIdx.x;
    const float t0 = theta[p * 3 + 0];
    const float t1 = theta[p * 3 + 1];
    const float t2 = theta[p * 3 + 2];
    const float* R = rot + (size_t)b * 9;
#pragma unroll
    for (int e = 0; e < 3; ++e) {
        proj[((size_t)b * NPROJ + p) * 3 + e] =
            t0 * R[e] + t1 * R[3 + e] + t2 * R[6 + e];
    }
}

// ---------------------------------------------------------------------------
// Kernel 2: xp[b][p][n] = sum_d x[b][n][d] * proj[b][p][d]  (same for yp)
// Batched GEMM via V_WMMA_F32_16X16X4_F32: M=16 projs, N=16 points, K=4
// (K=3 contraction zero-padded to 4).
// A (16x4 f32, 2 VGPR): lanes 0-15 hold {K=0,K=1}, lanes 16-31 hold {K=2,K=3}.
// B (4x16 f32, 2 VGPR): mirrored K striping, N across lanes.
// D (16x16 f32, 8 VGPR): VGPR r = row r (lanes 0-15) / row r+8 (lanes 16-31).
// Block = 256 threads = 8 waves; each wave owns 32 point-tiles.
// All loads are unconditional (per-lane computed offsets; hi lanes re-load
// element 2 as a dummy) so the hot loop has no EXEC-mask branching; the K=3
// zero-pad is a post-load v_cndmask.
// ---------------------------------------------------------------------------
__global__ void project_wmma_kernel(const float* __restrict__ x,
                                    const float* __restrict__ y,
                                    const float* __restrict__ proj,
                                    float* __restrict__ xp,
                                    float* __restrict__ yp,
                                    int b0) {
    const int cb    = blockIdx.x;          // chunk-local batch
    const int b     = b0 + cb;             // global batch
    const int pTile = blockIdx.y;          // 0..15
    const int wave  = threadIdx.x >> 5;
    const int lane  = threadIdx.x & 31;
    const int lm    = lane & 15;
    const bool hi   = lane >= 16;
    const int p0    = pTile * 16;

    // Per-lane element offsets: lanes 0-15 read {0,1}; lanes 16-31 read {2,2}
    // (second value discarded by the select below).
    const int o0 = hi ? 2 : 0;
    const int o1 = hi ? 2 : 1;

    // A-matrix: 16 projection rows, K = {p.x, p.y, p.z, 0}
    const float* prow = proj + ((size_t)b * NPROJ + p0 + lm) * 3;
    {
        // unconditional loads + select
        float u0 = prow[o0];
        float u1 = prow[o1];
        (void)u1;
    }
    float a0 = prow[o0];
    float a1 = prow[o1];
    v2f a;
    a.x = a0;
    a.y = hi ? 0.0f : a1;

    const float* xb  = x  + (size_t)b  * NPTS * 3;
    const float* yb  = y  + (size_t)b  * NPTS * 3;
    float*       xpc = xp + (size_t)cb * NPROJ * NPTS;
    float*       ypc = yp + (size_t)cb * NPROJ * NPTS;

    for (int i = 0; i < 32; ++i) {
        const int n0 = (wave * 32 + i) * 16;
        const float* xr = xb + (size_t)(n0 + lm) * 3;
        const float* yr = yb + (size_t)(n0 + lm) * 3;

        const float x0 = xr[o0];
        const float x1 = xr[o1];
        const float y0 = yr[o0];
        const float y1 = yr[o1];

        v2f bx, by;
        bx.x = x0;
        bx.y = hi ? 0.0f : x1;
        by.x = y0;
        by.y = hi ? 0.0f : y1;

        v8f cx = {};
        v8f cy = {};
        cx = __builtin_amdgcn_wmma_f32_16x16x4_f32(
                 false, a, false, bx, (short)0, cx, false, false);
        cy = __builtin_amdgcn_wmma_f32_16x16x4_f32(
                 false, a, false, by, (short)0, cy, false, false);

        const int rbase = hi ? 8 : 0;
#pragma unroll
        for (int r = 0; r < 8; ++r) {
            const size_t off = (size_t)(p0 + rbase + r) * NPTS + n0 + lm;
            xpc[off] = cx[r];
            ypc[off] = cy[r];
        }
    }
}

// ---------------------------------------------------------------------------
// Kernel 3: per (b,p): bitonic-sort xp row and yp row in LDS, then
// per_proj[b][p] = sum_n (sorted_x[n] - sorted_y[n])^2
// 512 threads, 32 KB LDS. Pair-index formulation: every inner iteration does
// a compare-exchange (2048 pairs / 512 threads = 4 iterations per pass).
// x & y are compare-exchanged in the same pass to amortize barriers.
// ---------------------------------------------------------------------------
__global__ void sortdiff_kernel(const float* __restrict__ xp,
                                const float* __restrict__ yp,
                                float* __restrict__ per_proj,
                                int b0) {
    __shared__ float sx[NPTS];
    __shared__ float sy[NPTS];
    const int cb  = blockIdx.x;
    const int p   = blockIdx.y;
    const int tid = threadIdx.x;
    const size_t base = ((size_t)cb * NPROJ + p) * NPTS;

    for (int i = tid; i < NPTS; i += 512) {
        sx[i] = xp[base + i];
        sy[i] = yp[base + i];
    }
    __syncthreads();

    for (int k = 2; k <= NPTS; k <<= 1) {
        for (int j = k >> 1; j > 0; j >>= 1) {
            const int jm1 = j - 1;
#pragma unroll 4
            for (int t = tid; t < NPTS / 2; t += 512) {
                const int i  = ((t & ~jm1) << 1) | (t & jm1);
                const int ix = i | j;
                const bool up = (i & k) == 0;
                float a0 = sx[i], a1 = sx[ix];
                if ((a0 > a1) == up) { sx[i] = a1; sx[ix] = a0; }
                float c0 = sy[i], c1 = sy[ix];
                if ((c0 > c1) == up) { sy[i] = c1; sy[ix] = c0; }
            }
            __syncthreads();
        }
    }

    float acc = 0.0f;
    for (int i = tid; i < NPTS; i += 512) {
        const float d = sx[i] - sy[i];
        acc += d * d;
    }
    __syncthreads();
    sx[tid] = acc;
    __syncthreads();
    for (int s = 256; s > 0; s >>= 1) {
        if (tid < s) sx[tid] += sx[tid + s];
        __syncthreads();
    }
    if (tid == 0) per_proj[(size_t)(b0 + cb) * NPROJ + p] = sx[0];
}

// ---------------------------------------------------------------------------
// Kernel 4: out = mean_b sqrt( mean_p per_proj[b][p] )    (one wave32)
// ---------------------------------------------------------------------------
__global__ void final_kernel(const float* __restrict__ per_proj,
                             float* __restrict__ out) {
    const int lane = threadIdx.x;  // 0..31 == batch index
    float s = 0.0f;
    for (int p = 0; p < NPROJ; ++p) s += per_proj[(size_t)lane * NPROJ + p];
    float dist = sqrtf(s * (1.0f / NPROJ));
#pragma unroll
    for (int off = 16; off > 0; off >>= 1)
        dist += __shfl_down(dist, off, 32);
    if (lane == 0) out[0] = dist * (1.0f / BATCH);
}

// ---------------------------------------------------------------------------
// Host launcher. Workspace layout:
//   [0, 98304)        proj    : 32*256*3 f32
//   [98304, 131072)   per_proj: 32*256 f32
//   [131072, ...)     xp/yp chunk buffers (8 MB per batch), chunked over
//                     batches so we fit whatever ws_size we were given.
// Chunk count depends only on ws_size -> deterministic launch sequence.
// ---------------------------------------------------------------------------
extern "C" void kernel_launch(void* const* d_in, const int* in_sizes, int n_in,
                              void* d_out, int out_size, void* d_ws, size_t ws_size,
                              hipStream_t stream) {
    (void)in_sizes; (void)n_in; (void)out_size;
    const float* x     = (const float*)d_in[0];
    const float* y     = (const float*)d_in[1];
    const float* theta = (const float*)d_in[2];
    const float* rot   = (const float*)d_in[3];
    float* out = (float*)d_out;

    char* ws = (char*)d_ws;
    float* proj     = (float*)ws;                 // 98304 bytes
    float* per_proj = (float*)(ws + 98304);       // 32768 bytes
    float* chunk    = (float*)(ws + 131072);

    const size_t perBatchBytes = (size_t)2 * NPROJ * NPTS * sizeof(float); // 8 MB
    const size_t avail = (ws_size > 131072) ? (ws_size - 131072) : 0;
    int C = (int)(avail / perBatchBytes);
    if (C < 1) C = 1;
    if (C > BATCH) C = BATCH;

    proj_kernel<<<BATCH, NPROJ, 0, stream>>>(theta, rot, proj);

    for (int b0 = 0; b0 < BATCH; b0 += C) {
        const int c = (BATCH - b0 < C) ? (BATCH - b0) : C;
        float* xpbuf = chunk;
        float* ypbuf = chunk + (size_t)c * NPROJ * NPTS;

        dim3 gp((unsigned)c, 16);
        project_wmma_kernel<<<gp, 256, 0, stream>>>(x, y, proj, xpbuf, ypbuf, b0);

        dim3 gs((unsigned)c, NPROJ);
        sortdiff_kernel<<<gs, 512, 0, stream>>>(xpbuf, ypbuf, per_proj, b0);
    }

    final_kernel<<<1, 32, 0, stream>>>(per_proj, out);
}